// GCN_12824772346537
// MI455X (gfx1250) — compile-verified
//
#include <hip/hip_runtime.h>
#include <cmath>
#include <cstdint>

#define D_IN  128
#define D_HID 64
#define D_OUT 16

typedef __attribute__((ext_vector_type(2))) float v2f;
typedef __attribute__((ext_vector_type(8))) float v8f;

// Raw LDS byte offset of a pointer into __shared__ memory (addrspace(3) ptrtoint).
__device__ __forceinline__ unsigned lds_offset(const void* p) {
  return (unsigned)(uintptr_t)(__attribute__((address_space(3))) const void*)p;
}

// ---------------- degree / symmetric norm ----------------
__global__ void k_deg_init(float* deg, int n) {
  int i = blockIdx.x * blockDim.x + threadIdx.x;
  if (i < n) deg[i] = 1.0f;               // self loop contributes 1
}

__global__ void k_deg_acc(const int* __restrict__ dst, float* deg, int e) {
  int i = blockIdx.x * blockDim.x + threadIdx.x;
  if (i < e) unsafeAtomicAdd(&deg[dst[i]], 1.0f);
}

__global__ void k_dinv(const float* __restrict__ deg, float* dinv, int n) {
  int i = blockIdx.x * blockDim.x + threadIdx.x;
  if (i < n) dinv[i] = rsqrtf(deg[i]);    // deg >= 1 always
}

// ---------------- f32 WMMA GEMM:  C[M,NOUT] = A[M,K] @ B[K,NOUT] ----------------
// Block = 8 waves; each wave computes one 16-row tile across all NOUT columns.
// LDS: B stored transposed [NOUT][K] (+4 pad) -> B-frag {B[k][n],B[k+1][n]} is one b64.
//      A tile per wave [16][K] (+4 pad), staged with GLOBAL_LOAD_ASYNC_TO_LDS_B128
//      (no VGPR round-trip; fenced with s_wait_asynccnt before fragment reads).
template<int K, int NOUT>
__global__ __launch_bounds__(256) void k_gemm_wmma(
    const float* __restrict__ A, const float* __restrict__ B,
    float* __restrict__ C, int mrows) {
  constexpr int NT  = NOUT / 16;
  constexpr int LDK = K + 4;              // padded row stride (floats), keeps banks spread
  constexpr int WAVES = 8;

  __shared__ float sB[NOUT * LDK];
  __shared__ float sA[WAVES * 16 * LDK];

  const int tid  = threadIdx.x;
  const int wave = tid >> 5;
  const int lane = tid & 31;

  const int tile = blockIdx.x * WAVES + wave;   // 16-row M tile index
  float* aw = &sA[wave * (16 * LDK)];
  const bool active = (tile * 16) < mrows;      // wave-uniform -> EXEC all-ones in WMMA

  // kick off async A-tile staging first so it overlaps the B transpose below
  if (active) {
    const float* arow = A + (size_t)tile * (16 * K);
    const unsigned awbase = lds_offset(aw);
    for (int i = lane; i < (16 * K) / 4; i += 32) {
      int r = (i * 4) / K, c = (i * 4) % K;     // 16B chunk stays within one padded row
      unsigned loff = awbase + (unsigned)((r * LDK + c) * sizeof(float));
      unsigned long long ga = (unsigned long long)(uintptr_t)(arow + r * K + c);
      asm volatile("global_load_async_to_lds_b128 %0, %1, off"
                   :: "v"(loff), "v"(ga) : "memory");
    }
  }

  // stage B transposed (B is [K][NOUT] row-major), coalesced global reads
  for (int idx = tid; idx < K * NOUT; idx += 256) {
    int k = idx / NOUT, nn = idx % NOUT;
    sB[nn * LDK + k] = B[idx];
  }

  if (active) {
    asm volatile("s_wait_asynccnt 0" ::: "memory");  // A tile resident (wave-private)
  }
  __syncthreads();                                    // sB visible block-wide
  if (!active) return;

  v8f acc[NT] = {};
  const int r  = lane & 15;                 // M row (A) / N column (B,C)
  const int ko = (lane >> 4) << 1;          // half-wave K offset: 0 or 2

  for (int ks = 0; ks < K / 4; ++ks) {
    const int k = ks * 4 + ko;
    const v2f af = *(const v2f*)&aw[r * LDK + k];       // {A[r][k], A[r][k+1]}
#pragma unroll
    for (int t = 0; t < NT; ++t) {
      const v2f bf = *(const v2f*)&sB[(t * 16 + r) * LDK + k]; // {B[k][n], B[k+1][n]}
      acc[t] = __builtin_amdgcn_wmma_f32_16x16x4_f32(
          false, af, false, bf, (short)0, acc[t], false, false);
    }
  }

  // C/D layout: VGPR g -> row g (lanes 0-15) / row g+8 (lanes 16-31), N = lane&15
  const int mbase = tile * 16 + ((lane >> 4) << 3);
#pragma unroll
  for (int g = 0; g < 8; ++g) {
    float* crow = C + (size_t)(mbase + g) * NOUT + r;
#pragma unroll
    for (int t = 0; t < NT; ++t) crow[t * 16] = acc[t][g];
  }
}

// ---------------- edge gather-scale-scatter: out[dst] += h[src] * dinv[s]*dinv[d] ----------------
template<int D>
__global__ void k_edge_agg(const int* __restrict__ src, const int* __restrict__ dst,
                           const float* __restrict__ dinv, const float* __restrict__ h,
                           float* out, int e) {
  constexpr int C = D / 4;
  long long idx = (long long)blockIdx.x * blockDim.x + threadIdx.x;
  if (idx >= (long long)e * C) return;
  int ei = (int)(idx / C);
  int c  = (int)(idx % C);
  int s = src[ei], d = dst[ei];
  float w = dinv[s] * dinv[d];
  const float4 v = *(const float4*)&h[(size_t)s * D + c * 4];
  float* o = out + (size_t)d * D + c * 4;
  unsafeAtomicAdd(o + 0, v.x * w);
  unsafeAtomicAdd(o + 1, v.y * w);
  unsafeAtomicAdd(o + 2, v.z * w);
  unsafeAtomicAdd(o + 3, v.w * w);
}

// ---------------- a1 = sigmoid(agg1 + self_loop + b1) (in place over agg1) ----------------
__global__ void k_post1(float* agg, const float* __restrict__ h1,
                        const float* __restrict__ dinv, const float* __restrict__ b1, int n) {
  int i = blockIdx.x * blockDim.x + threadIdx.x;
  if (i >= n * D_HID) return;
  int node = i >> 6, f = i & (D_HID - 1);
  float di = dinv[node];
  float z = agg[i] + h1[i] * di * di + b1[f];
  agg[i] = 1.0f / (1.0f + expf(-z));
}

// ---------------- out = log_softmax(agg2 + self_loop + b2) ----------------
__global__ void k_final(const float* __restrict__ agg2, const float* __restrict__ h2,
                        const float* __restrict__ dinv, const float* __restrict__ b2,
                        float* __restrict__ out, int n) {
  int i = blockIdx.x * blockDim.x + threadIdx.x;
  if (i >= n) return;
  float dd = dinv[i] * dinv[i];
  float z[D_OUT];
  float m = -INFINITY;
#pragma unroll
  for (int f = 0; f < D_OUT; ++f) {
    z[f] = agg2[i * D_OUT + f] + h2[i * D_OUT + f] * dd + b2[f];
    m = fmaxf(m, z[f]);
  }
  float s = 0.0f;
#pragma unroll
  for (int f = 0; f < D_OUT; ++f) s += expf(z[f] - m);
  float lse = logf(s);
#pragma unroll
  for (int f = 0; f < D_OUT; ++f) out[i * D_OUT + f] = z[f] - m - lse;
}

extern "C" void kernel_launch(void* const* d_in, const int* in_sizes, int n_in,
                              void* d_out, int out_size, void* d_ws, size_t ws_size,
                              hipStream_t stream) {
  const float* x  = (const float*)d_in[0];
  const int*   ei = (const int*)d_in[1];
  const float* W1 = (const float*)d_in[2];
  const float* b1 = (const float*)d_in[3];
  const float* W2 = (const float*)d_in[4];
  const float* b2 = (const float*)d_in[5];
  float* out = (float*)d_out;

  const int n = in_sizes[0] / D_IN;      // 100000
  const int e = in_sizes[1] / 2;         // 1600000
  const int* src = ei;
  const int* dst = ei + e;

  // workspace layout (floats): deg | dinv | h1[n*64] | agg1[n*64]
  // h1 region is recycled after k_post1: h2[n*16] and agg2[n*16] live inside it.
  float* ws   = (float*)d_ws;
  float* deg  = ws;
  float* dinv = deg + n;
  float* h1   = dinv + n;
  float* agg1 = h1 + (size_t)n * D_HID;  // becomes a1 after k_post1
  float* h2   = h1;                       // reuse (h1 dead after k_post1)
  float* agg2 = h1 + (size_t)n * D_OUT;

  const int B = 256;
  k_deg_init<<<(n + B - 1) / B, B, 0, stream>>>(deg, n);
  k_deg_acc<<<(e + B - 1) / B, B, 0, stream>>>(dst, deg, e);
  k_dinv<<<(n + B - 1) / B, B, 0, stream>>>(deg, dinv, n);

  const int tiles   = (n + 15) / 16;     // 6250 (exact)
  const int gblocks = (tiles + 7) / 8;

  // layer 1
  k_gemm_wmma<D_IN, D_HID><<<gblocks, 256, 0, stream>>>(x, W1, h1, n);
  hipMemsetAsync(agg1, 0, (size_t)n * D_HID * sizeof(float), stream);
  {
    long long total = (long long)e * (D_HID / 4);
    k_edge_agg<D_HID><<<(unsigned)((total + B - 1) / B), B, 0, stream>>>(
        src, dst, dinv, h1, agg1, e);
  }
  k_post1<<<((n * D_HID) + B - 1) / B, B, 0, stream>>>(agg1, h1, dinv, b1, n);

  // layer 2
  k_gemm_wmma<D_HID, D_OUT><<<gblocks, 256, 0, stream>>>(agg1, W2, h2, n);
  hipMemsetAsync(agg2, 0, (size_t)n * D_OUT * sizeof(float), stream);
  {
    long long total = (long long)e * (D_OUT / 4);
    k_edge_agg<D_OUT><<<(unsigned)((total + B - 1) / B), B, 0, stream>>>(
        src, dst, dinv, h2, agg2, e);
  }
  k_final<<<(n + B - 1) / B, B, 0, stream>>>(agg2, h2, dinv, b2, out, n);
}